// Model_21337397527223
// MI455X (gfx1250) — compile-verified
//
#include <hip/hip_runtime.h>
#include <stdint.h>

// Memory-bound streaming kernel: ~394 MB traffic -> ~17us roofline @ 23.3 TB/s.
// No matmul structure (min/max overlaps) -> WMMA not applicable. CDNA5 path:
// async global->LDS staging (GVS mode: SGPR base + v_off + inst offset, NT hint),
// s_wait_asynccnt, then 4x4 min/max reductions out of LDS.

#define BLK 256
#define PRED_F 30          // floats per pred element (10 boxes x 3), only first 12 used
#define TRUE_F 16          // floats per true element
#define TRUE_STRIDE_F 20   // padded LDS stride: 80B (16B-aligned b128 dest, <=2-way bank conflict)

// GVS-form async copy: copies 16 bytes from global memory at sbase plus voff to LDS at lds_byte, each offset by IOFF
// (per CDNA5 ISA, INST_OFFSET is added to BOTH addresses).
template <int IOFF>
__device__ __forceinline__ void async_g2lds_b128(uint32_t lds_byte, uint32_t voff,
                                                 const void* sbase) {
  asm volatile("global_load_async_to_lds_b128 %0, %1, %2 offset:%3 th:TH_LOAD_NT"
               :: "v"(lds_byte), "v"(voff), "s"(sbase), "i"(IOFF)
               : "memory");
}

__device__ __forceinline__ void s_wait_async0() {
  asm volatile("s_wait_asynccnt 0x0" ::: "memory");
}

__device__ __forceinline__ float ov(float lo1, float hi1, float lo2, float hi2) {
  return fmaxf(0.0f, fminf(hi1, hi2) - fmaxf(lo1, lo2));
}

__global__ void __launch_bounds__(BLK)
beta_iou_loss_kernel(const float* __restrict__ tr, const float* __restrict__ pr,
                     float* __restrict__ out, int n) {
  __shared__ __align__(16) float s_true[BLK * TRUE_STRIDE_F]; // 20 KB
  __shared__ __align__(16) float s_pred[BLK * PRED_F];        // 30 KB

  const int t  = threadIdx.x;
  const int bb = blockIdx.x * BLK;
  int rem = n - bb;
  if (rem > BLK) rem = BLK;          // uniform across block

  const uint32_t ldsT = (uint32_t)(uintptr_t)s_true; // low 32 bits = LDS byte addr
  const uint32_t ldsP = (uint32_t)(uintptr_t)s_pred;

  // Uniform SGPR bases (block offset folded in; both 16B-aligned: 64*bb, 120*bb=30720*blk).
  const char* trBase = (const char*)tr + (size_t)bb * (TRUE_F * 4);
  const char* prBase = (const char*)pr + (size_t)bb * (PRED_F * 4);

  const uint32_t voff  = (uint32_t)t * 16u;                                   // shared v-offset
  const uint32_t vdstT = ldsT + (uint32_t)((t >> 2) * (TRUE_STRIDE_F * 4) + (t & 3) * 16);
  const uint32_t vdstP = ldsP + voff;

  // ---- stage TRUE: rem*64B as 16B chunks; chunk (t,k): glb += 4096k, LDS += 5120k
  //      (4096k via INST_OFFSET on both sides, remaining 1024k folded into VDST) ----
  const int chunksT = rem * 4;
#define STAGE_T(K)                                                   \
  if (t + (K) * BLK < chunksT)                                       \
    async_g2lds_b128<(K) * 4096>(vdstT + (K) * 1024, voff, trBase);
  STAGE_T(0) STAGE_T(1) STAGE_T(2) STAGE_T(3)
#undef STAGE_T

  // ---- stage PRED: rem*30 floats contiguous as 16B chunks; both sides advance 4096k ----
  const int totF = rem * PRED_F;
  const int vecP = totF >> 2;        // full 16B chunks (1920 for a full block)
#define STAGE_P(K)                                                   \
  if (t + (K) * BLK < vecP)                                          \
    async_g2lds_b128<(K) * 4096>(vdstP, voff, prBase);
  STAGE_P(0) STAGE_P(1) STAGE_P(2) STAGE_P(3)
  STAGE_P(4) STAGE_P(5) STAGE_P(6) STAGE_P(7)
#undef STAGE_P
  // scalar tail (only when rem odd; never taken for B multiple of 256)
  for (int i = (vecP << 2) + t; i < totF; i += BLK)
    s_pred[i] = pr[(size_t)bb * PRED_F + i];

  s_wait_async0();     // this wave's async copies into LDS complete
  __syncthreads();     // all waves' copies visible

  if (t >= rem) return;

  // ---- unpack true boxes (16B-aligned ds_load_b128, padded rows: <=2-way conflicts) ----
  float ix1[4], iy1[4], ax1[4], ay1[4];
  const float* T = s_true + t * TRUE_STRIDE_F;
#pragma unroll
  for (int i = 0; i < 4; ++i) {
    const float4 b = *reinterpret_cast<const float4*>(T + 4 * i);
    ix1[i] = b.x; iy1[i] = b.y; ax1[i] = b.z; ay1[i] = b.w;
  }

  // ---- unpack + canonicalize pred boxes (first 4 of 10; stride 30 conflict-free in wave32) ----
  float ix2[4], iy2[4], ax2[4], ay2[4];
  const float* P = s_pred + t * PRED_F;
#pragma unroll
  for (int j = 0; j < 4; ++j) {
    const float a = P[3 * j + 0];
    const float y = P[3 * j + 1];
    const float b = P[3 * j + 2];
    const float lo = fminf(a, b), hi = fmaxf(a, b);
    ix2[j] = lo; ax2[j] = hi;
    iy2[j] = y;  ay2[j] = y + (hi - lo);
  }

  // ---- 4x4 reductions: intersection + both self-betas (diag area squared) ----
  float inter = 0.0f, beta1 = 0.0f, beta2 = 0.0f;
#pragma unroll
  for (int i = 0; i < 4; ++i) {
#pragma unroll
    for (int j = 0; j < 4; ++j) {
      inter += ov(ix1[i], ax1[i], ix2[j], ax2[j]) * ov(iy1[i], ay1[i], iy2[j], ay2[j]);
      {
        const float a = ov(ix1[i], ax1[i], ix1[j], ax1[j]) * ov(iy1[i], ay1[i], iy1[j], ay1[j]);
        beta1 += (i == j) ? a * a : a;
      }
      {
        const float a = ov(ix2[i], ax2[i], ix2[j], ax2[j]) * ov(iy2[i], ay2[i], iy2[j], ay2[j]);
        beta2 += (i == j) ? a * a : a;
      }
    }
  }
  const float d = beta1 - beta2;
  __builtin_nontemporal_store(d * d - inter, &out[bb + t]);  // coalesced NT b32 store
}

extern "C" void kernel_launch(void* const* d_in, const int* in_sizes, int n_in,
                              void* d_out, int out_size, void* d_ws, size_t ws_size,
                              hipStream_t stream) {
  (void)in_sizes; (void)n_in; (void)d_ws; (void)ws_size;
  const float* tr = (const float*)d_in[0];  // true: (B,4,4) float32
  const float* pr = (const float*)d_in[1];  // pred: (B,10,3) float32
  float* out = (float*)d_out;               // (B,1) float32
  const int n = out_size;
  const int blocks = (n + BLK - 1) / BLK;
  beta_iou_loss_kernel<<<blocks, BLK, 0, stream>>>(tr, pr, out, n);
}